// MoERouter_47914655154651
// MI455X (gfx1250) — compile-verified
//
#include <hip/hip_runtime.h>
#include <hip/hip_bf16.h>

// MoE softmax router for MI455X (gfx1250, wave32, WMMA).
//
// Fused: logits GEMM (f32 WMMA 16x16x4) + softmax(64) + top-2 + dispatch.
// Memory-bound (x stream = 134 MB @ 23.3 TB/s ~ 6 us), so f32 precision is
// free; logits never leave registers.
//
// Round-2 change: W is staged into LDS in K-pair-interleaved layout
//   sW[p][e][2] = { W[2p][e], W[2p+1][e] }
// so each WMMA B fragment is a single contiguous 8B ds_load_b64 into an even
// VGPR pair (no v_dual_mov repacking, fewer dscnt waits in the hot loop).

typedef __attribute__((ext_vector_type(2))) float v2f;
typedef __attribute__((ext_vector_type(8))) float v8f;

#define D_MODEL   2048
#define N_EXPERTS 64
#define KCHUNK    128           // W staging chunk: 128 x 64 f32 = 32 KB LDS
#define WAVES_PB  4             // 4 waves/block -> 128 threads, 64 tokens/block
#define NTHREADS  (WAVES_PB * 32)

__global__ __launch_bounds__(NTHREADS)
void moe_router_wmma_f32(const float* __restrict__ x,
                         const float* __restrict__ W,
                         float* __restrict__ out_dispatch,
                         float* __restrict__ out_probs,
                         float* __restrict__ out_tkp,
                         float* __restrict__ out_tki)
{
    // K-pair interleaved: sW[p*128 + e*2 + r] = W[kc + 2p + r][e], p in [0,64)
    __shared__ float sW[KCHUNK * N_EXPERTS];    // 32 KB

    const int tid  = threadIdx.x;
    const int wave = tid >> 5;
    const int lane = tid & 31;
    const int h    = lane >> 4;                 // lane-half: 0 or 1
    const int le   = lane & 15;                 // lane within half

    // each wave owns a 16-token tile; computes all 64 experts (4 N-tiles)
    const int tile      = blockIdx.x * WAVES_PB + wave;
    const int tokenBase = tile * 16;

    // A-operand row for this lane (ISA 7.12.2: 32-bit A 16x4, M = lane%16)
    const float* xrow = x + (size_t)(tokenBase + le) * D_MODEL;

    v8f acc[4];
#pragma unroll
    for (int t = 0; t < 4; ++t) acc[t] = (v8f){0.f,0.f,0.f,0.f,0.f,0.f,0.f,0.f};

    for (int kc = 0; kc < D_MODEL; kc += KCHUNK) {
        __syncthreads();
        // --- stage + swizzle W[kc .. kc+KCHUNK) into pair-interleaved LDS ---
        // unit u = (pair p, expert-group eg of 4): 64 pairs * 16 groups = 1024
        {
            float4* sW4 = (float4*)sW;
#pragma unroll
            for (int i = 0; i < (KCHUNK / 2) * 16 / NTHREADS; ++i) {
                const int u  = tid + i * NTHREADS;
                const int eg = u & 15;
                const int p  = u >> 4;
                const float4 r0 = *(const float4*)(W + (size_t)(kc + 2*p    ) * N_EXPERTS + 4*eg);
                const float4 r1 = *(const float4*)(W + (size_t)(kc + 2*p + 1) * N_EXPERTS + 4*eg);
                sW4[p * 32 + eg * 2 + 0] = (float4){r0.x, r1.x, r0.y, r1.y};
                sW4[p * 32 + eg * 2 + 1] = (float4){r0.z, r1.z, r0.w, r1.w};
            }
        }
        // keep the x stream warm across the staging barrier (global_prefetch_b8)
        if (kc + KCHUNK < D_MODEL)
            __builtin_prefetch(xrow + kc + KCHUNK + 2 * h, 0, 0);
        __syncthreads();

#pragma unroll 4
        for (int kl = 0; kl < KCHUNK; kl += 4) {
            // A fragment: lanes 0-15 carry K=kl,kl+1; lanes 16-31 carry K=kl+2,kl+3
            v2f a = *(const v2f*)(xrow + kc + kl + 2 * h);
            // B fragments: contiguous b64 per tile, imm offsets 0/128/256/384 B
            const v2f* brow = (const v2f*)&sW[(kl / 2 + h) * (2 * N_EXPERTS)];
#pragma unroll
            for (int t = 0; t < 4; ++t) {
                v2f b = brow[16 * t + le];
                // D = A(16x4,f32) * B(4x16,f32) + C  -> v_wmma_f32_16x16x4_f32
                acc[t] = __builtin_amdgcn_wmma_f32_16x16x4_f32(
                    /*neg_a=*/false, a, /*neg_b=*/false, b,
                    /*c_mod=*/(short)0, acc[t],
                    /*reuse_a=*/false, /*reuse_b=*/false);
            }
        }
    }

    // ---- epilogue: softmax over 64 experts, top-2, dispatch --------------
    // C layout: vgpr j, lane l  ->  token = tokenBase + j + 8*h, expert = 16t+le.
    // Each 16-lane half holds 8 full token rows -> width-16 butterflies reduce rows.
#pragma unroll
    for (int j = 0; j < 8; ++j) {
        const int token = tokenBase + j + 8 * h;

        float p0 = acc[0][j], p1 = acc[1][j], p2 = acc[2][j], p3 = acc[3][j];

        // row max (4 local + butterfly over the 16-lane half)
        float mx = fmaxf(fmaxf(p0, p1), fmaxf(p2, p3));
#pragma unroll
        for (int m = 1; m < 16; m <<= 1) mx = fmaxf(mx, __shfl_xor(mx, m, 16));

        float e0 = __expf(p0 - mx), e1 = __expf(p1 - mx);
        float e2 = __expf(p2 - mx), e3 = __expf(p3 - mx);
        float s = (e0 + e1) + (e2 + e3);
#pragma unroll
        for (int m = 1; m < 16; m <<= 1) s += __shfl_xor(s, m, 16);
        const float inv = 1.0f / s;

        float pr[4] = { e0 * inv, e1 * inv, e2 * inv, e3 * inv };

        // router_probs: lanes 0-15 of a half write 64B contiguous per t
        float* prow = out_probs + (size_t)token * N_EXPERTS;
#pragma unroll
        for (int t = 0; t < 4; ++t) prow[16 * t + le] = pr[t];

        // per-lane top-2 of its 4 experts (lower index wins ties)
        float v1 = -1.0f, v2 = -1.0f;   // probs are in (0,1): -1 is safe -inf
        int   i1 = 0,     i2 = 0;
#pragma unroll
        for (int t = 0; t < 4; ++t) {
            const int e = 16 * t + le;
            if (pr[t] > v1)      { v2 = v1; i2 = i1; v1 = pr[t]; i1 = e; }
            else if (pr[t] > v2) { v2 = pr[t]; i2 = e; }
        }
        // butterfly merge of sorted pairs across the 16-lane half
#pragma unroll
        for (int m = 1; m < 16; m <<= 1) {
            float ov1 = __shfl_xor(v1, m, 16), ov2 = __shfl_xor(v2, m, 16);
            int   oi1 = __shfl_xor(i1, m, 16), oi2 = __shfl_xor(i2, m, 16);
            bool mineFirst = (v1 > ov1) || (v1 == ov1 && i1 < oi1);
            float nv1, nv2; int ni1, ni2;
            if (mineFirst) {
                nv1 = v1; ni1 = i1;
                bool s2 = (v2 > ov1) || (v2 == ov1 && i2 < oi1);
                nv2 = s2 ? v2 : ov1;  ni2 = s2 ? i2 : oi1;
            } else {
                nv1 = ov1; ni1 = oi1;
                bool s2 = (ov2 > v1) || (ov2 == v1 && oi2 < i1);
                nv2 = s2 ? ov2 : v1;  ni2 = s2 ? oi2 : i1;
            }
            v1 = nv1; i1 = ni1; v2 = nv2; i2 = ni2;
        }

        // dispatch one-hot sum (all lanes know the final top-2)
        float* drow = out_dispatch + (size_t)token * N_EXPERTS;
#pragma unroll
        for (int t = 0; t < 4; ++t) {
            const int e = 16 * t + le;
            drow[e] = (e == i1 || e == i2) ? 1.0f : 0.0f;
        }

        if (le == 0) {
            out_tkp[(size_t)token * 2 + 0] = v1;
            out_tkp[(size_t)token * 2 + 1] = v2;
            out_tki[(size_t)token * 2 + 0] = (float)i1;
            out_tki[(size_t)token * 2 + 1] = (float)i2;
        }
    }
}

extern "C" void kernel_launch(void* const* d_in, const int* in_sizes, int n_in,
                              void* d_out, int out_size, void* d_ws, size_t ws_size,
                              hipStream_t stream) {
    const float* x = (const float*)d_in[0];      // [B,S,D] f32
    const float* W = (const float*)d_in[1];      // [D,E]   f32

    const int n_tokens = in_sizes[0] / D_MODEL;  // 16384

    float* out      = (float*)d_out;
    float* dispatch = out;                                        // [N,64]
    float* probs    = out + (size_t)n_tokens * N_EXPERTS;         // [N,64]
    float* tkp      = out + (size_t)n_tokens * N_EXPERTS * 2;     // [N,2]
    float* tki      = tkp + (size_t)n_tokens * 2;                 // [N,2] (as f32)

    const int tiles  = n_tokens / 16;            // 1024 wave-tiles
    const int blocks = tiles / WAVES_PB;         // 256 blocks of 128 threads

    moe_router_wmma_f32<<<blocks, NTHREADS, 0, stream>>>(
        x, W, dispatch, probs, tkp, tki);
}